// AdaptiveWaveletLayerSparse_71055938945734
// MI455X (gfx1250) — compile-verified
//
#include <hip/hip_runtime.h>
#include <hip/hip_bf16.h>

// Problem constants (fixed by the reference's setup_inputs)
#define BB 8LL
#define NN 4096LL
#define TT 24LL
#define CC 64LL
#define EE 131072LL
#define GG (BB * TT)          // 192
#define NODES (GG * NN)       // 786432
#define TOT (GG * NN * CC)    // 50331648
#define EDGEI (GG * EE)       // 25165824
#define NEG_SLOPE 0.2f

typedef __attribute__((ext_vector_type(2))) float v2f;
typedef __attribute__((ext_vector_type(8))) float v8f;

// Offset of element (g, n, c=0) in either the strided x layout (B,N,T,C with
// g = b*T + t) or the contiguous working layout (g, n, c).
__device__ __forceinline__ long long cur_off(long long g, long long n, int xstrided) {
    if (xstrided) {
        long long b = g / TT;
        long long t = g - b * TT;
        return ((b * NN + n) * TT + t) * CC;
    }
    return (g * NN + n) * CC;
}

// ---- tiny: sigmoid of the gating scalars --------------------------------
__global__ void coef_kernel(const float* __restrict__ temp,
                            const float* __restrict__ cheb,
                            float* __restrict__ coefs) {
    if (threadIdx.x == 0 && blockIdx.x == 0) {
        coefs[0] = 1.f / (1.f + __expf(-temp[0]));
        coefs[1] = 1.f / (1.f + __expf(-temp[1]));
        coefs[2] = 1.f / (1.f + __expf(-temp[2]));
        coefs[3] = 1.f / (1.f + __expf(-cheb[0]));   // cheb_coe[0]
        coefs[4] = 1.f / (1.f + __expf(-cheb[1]));   // cheb_coe[1]
    }
}

// ---- f1[g,n] = cur[g,n,:].a1 ; f2[g,n] = cur[g,n,:].a2 ------------------
// One wave32 per node-instance; lane l handles channels 2l,2l+1 (coalesced
// 256B read), shuffle-reduce across the wave.
__global__ __launch_bounds__(256) void feat_kernel(const float* __restrict__ cur,
                                                   int xstrided,
                                                   const float* __restrict__ a,
                                                   float* __restrict__ f1,
                                                   float* __restrict__ f2) {
    long long wid = (((long long)blockIdx.x * blockDim.x) + threadIdx.x) >> 5;
    int lane = threadIdx.x & 31;
    if (wid >= NODES) return;
    long long g = wid >> 12;            // / NN
    long long n = wid & (NN - 1);
    const float* p = cur + cur_off(g, n, xstrided) + 2 * lane;
    v2f xv = *(const v2f*)p;
    v2f a1 = *(const v2f*)(a + 2 * lane);
    v2f a2 = *(const v2f*)(a + CC + 2 * lane);
    float s1 = xv.x * a1.x + xv.y * a1.y;
    float s2 = xv.x * a2.x + xv.y * a2.y;
#pragma unroll
    for (int off = 16; off > 0; off >>= 1) {
        s1 += __shfl_xor(s1, off, 32);
        s2 += __shfl_xor(s2, off, 32);
    }
    if (lane == 0) {
        f1[wid] = s1;
        f2[wid] = s2;
    }
}

// ---- softmax denominator: one thread per (g,e) --------------------------
__global__ __launch_bounds__(256) void edge_denom_kernel(const float* __restrict__ f1,
                                                         const float* __restrict__ f2,
                                                         const int* __restrict__ ei,
                                                         float* __restrict__ denom) {
    long long id = ((long long)blockIdx.x * blockDim.x) + threadIdx.x;
    if (id >= EDGEI) return;
    long long g = id >> 17;             // / EE
    long long e = id & (EE - 1);
    int dst = ei[e];
    int src = ei[EE + e];
    float v = f1[g * NN + dst] + f2[g * NN + src];
    v = (v >= 0.f) ? v : NEG_SLOPE * v;
    atomicAdd(&denom[g * NN + dst], __expf(v));
}

// ---- message aggregation: one wave32 per (g,e) --------------------------
// Lane l carries channels 2l,2l+1: coalesced 256B gather of cur[g,src,:],
// 64 consecutive f32 atomic adds into out[g,dst,:]. exp(e_ij) recomputed
// (f1/f2/denom are L2-resident) instead of storing a 100MB buffer.
__global__ __launch_bounds__(256) void edge_agg_kernel(const float* __restrict__ cur,
                                                       int xstrided,
                                                       const float* __restrict__ f1,
                                                       const float* __restrict__ f2,
                                                       const float* __restrict__ denom,
                                                       const int* __restrict__ ei,
                                                       float* __restrict__ out) {
    long long wid = (((long long)blockIdx.x * blockDim.x) + threadIdx.x) >> 5;
    int lane = threadIdx.x & 31;
    if (wid >= EDGEI) return;
    long long g = wid >> 17;
    long long e = wid & (EE - 1);
    int dst = ei[e];
    int src = ei[EE + e];
    float v = f1[g * NN + dst] + f2[g * NN + src];
    v = (v >= 0.f) ? v : NEG_SLOPE * v;
    float d = denom[g * NN + dst];
    d = (d > 1e-8f) ? d : 1e-8f;
    float alpha = __expf(v) / d;
    const float* sp = cur + cur_off(g, src, xstrided) + 2 * lane;
    float* dp = out + (g * NN + dst) * CC + 2 * lane;
    v2f sv = *(const v2f*)sp;
    atomicAdd(dp + 0, alpha * sv.x);
    atomicAdd(dp + 1, alpha * sv.y);
}

// ---- wavelet combine (elementwise, matches reference mutation order) ----
__global__ __launch_bounds__(256) void combine_kernel(const float* __restrict__ x,
                                                      const float* __restrict__ nxt,
                                                      float* __restrict__ fp,
                                                      float* __restrict__ upd,
                                                      const float* __restrict__ coefs,
                                                      int step) {
    long long i = ((long long)blockIdx.x * blockDim.x) + threadIdx.x;
    if (i >= TOT) return;
    long long c = i & (CC - 1);
    long long gn = i >> 6;
    long long n = gn & (NN - 1);
    long long g = gn >> 12;
    long long b = g / TT;
    long long t = g - b * TT;
    float xv = x[((b * NN + n) * TT + t) * CC + c];
    float u = nxt[i];
    float coe0 = coefs[0], coe1 = coefs[1], coe2 = coefs[2];
    float fe = coe0 * xv + u;              // feat_even
    if (step >= 1) u *= coefs[3 + (step - 1)]; // update = cheb_coe[step-1]*update
    float fo = u - fe;                     // feat_odd
    float ff = coe1 * fe + (1.f - coe1) * fo;  // feat_fuse
    float prev = (step == 0) ? xv : fp[i];
    fp[i] = coe2 * prev + (1.f - coe2) * ff;
    upd[i] = u;                            // input to next hop's attention
}

// ---- final GEMM with V_WMMA_F32_16X16X4_F32 + fused bias & transpose ----
// Block = 256 threads = 8 waves; block covers 32 rows x 64 cols.
// Wave w: row tile (w>>2)*16, col tile (w&3)*16; K=64 via 16 chained WMMAs.
// A 16x4 f32 layout: lane l holds row m=l%16, k = (l/16)*2 + {0,1}.
// D 16x16 f32 layout: lane l, VGPR v -> row v + 8*(l/16), col l%16.
__global__ __launch_bounds__(256) void gemm_wmma_kernel(const float* __restrict__ fp,
                                                        const float* __restrict__ Wm,
                                                        const float* __restrict__ bias,
                                                        float* __restrict__ out) {
    const int lane = threadIdx.x & 31;
    const int wave = threadIdx.x >> 5;
    const long long rowBase = (long long)blockIdx.x * 32 + (long long)(wave >> 2) * 16;
    const int colBase = (wave & 3) * 16;
    const int mrow = lane & 15;
    const int khalf = lane >> 4;

    v8f acc = {};
    const float* ap = fp + (rowBase + mrow) * CC + khalf * 2;
    const float* bp = Wm + (long long)(khalf * 2) * CC + (colBase + mrow);
#pragma unroll
    for (int kk = 0; kk < 16; ++kk) {
        v2f av = *(const v2f*)(ap + kk * 4);
        v2f bv;
        bv.x = bp[kk * 4 * CC];
        bv.y = bp[kk * 4 * CC + CC];
        acc = __builtin_amdgcn_wmma_f32_16x16x4_f32(
            /*neg_a=*/false, av, /*neg_b=*/false, bv,
            /*c_mod=*/(short)0, acc, /*reuse_a=*/false, /*reuse_b=*/false);
    }

    const int ncol = colBase + mrow;
    const float bb = bias[ncol];
    // Row tile never crosses a g boundary (16 | NN), so g,b,t constant here.
    const long long g = rowBase >> 12;
    const long long nBase = rowBase & (NN - 1);
    const long long b = g / TT;
    const long long t = g - b * TT;
#pragma unroll
    for (int v = 0; v < 8; ++v) {
        long long nrow = nBase + v + 8 * khalf;
        out[((b * NN + nrow) * TT + t) * CC + ncol] = acc[v] + bb;
    }
}

extern "C" void kernel_launch(void* const* d_in, const int* in_sizes, int n_in,
                              void* d_out, int out_size, void* d_ws, size_t ws_size,
                              hipStream_t stream) {
    const float* x     = (const float*)d_in[0];   // (B,N,T,C)
    // d_in[1] = h0 : unused by the reference
    const int*   edge  = (const int*)d_in[2];     // (2,E)
    const float* a     = (const float*)d_in[3];   // (2C,1)
    const float* temp  = (const float*)d_in[4];   // (4,)
    const float* cheb  = (const float*)d_in[5];   // (4,)
    const float* Wm    = (const float*)d_in[6];   // (C,Cout)
    const float* bias  = (const float*)d_in[7];   // (Cout,)
    float* out = (float*)d_out;                   // (B,N,T,C)

    float* ws    = (float*)d_ws;
    float* upd   = ws;                 // TOT : current update (attn input)
    float* nxt   = upd + TOT;          // TOT : attn output
    float* fp    = nxt + TOT;          // TOT : feat_prime accumulator
    float* f1    = fp + TOT;           // NODES
    float* f2    = f1 + NODES;         // NODES
    float* denom = f2 + NODES;         // NODES
    float* coefs = denom + NODES;      // 8

    coef_kernel<<<1, 32, 0, stream>>>(temp, cheb, coefs);

    const int blk = 256;
    const long long featBlocks  = NODES / 8;       // 1 wave per node, 8/block
    const long long denomBlocks = EDGEI / blk;     // exact
    const long long aggBlocks   = EDGEI / 8;       // 1 wave per edge-instance
    const long long combBlocks  = TOT / blk;       // exact

    for (int h = 0; h < 3; ++h) {
        const float* cur = (h == 0) ? x : upd;
        int xs = (h == 0) ? 1 : 0;

        feat_kernel<<<(int)featBlocks, blk, 0, stream>>>(cur, xs, a, f1, f2);
        hipMemsetAsync(denom, 0, NODES * sizeof(float), stream);
        edge_denom_kernel<<<(int)denomBlocks, blk, 0, stream>>>(f1, f2, edge, denom);
        hipMemsetAsync(nxt, 0, TOT * sizeof(float), stream);
        edge_agg_kernel<<<(int)aggBlocks, blk, 0, stream>>>(cur, xs, f1, f2, denom, edge, nxt);
        combine_kernel<<<(int)combBlocks, blk, 0, stream>>>(x, nxt, fp, upd, coefs, h);
    }

    gemm_wmma_kernel<<<(int)(NODES / 32), blk, 0, stream>>>(fp, Wm, bias, out);
}